// GraphConvolution_73933567033883
// MI455X (gfx1250) — compile-verified
//
#include <hip/hip_runtime.h>

typedef __attribute__((ext_vector_type(2))) float v2f;
typedef __attribute__((ext_vector_type(8))) float v8f;

#define F_IN  512
#define F_OUT 256

// ---------------------------------------------------------------------------
// Kernel 1: support = X @ W using V_WMMA_F32_16X16X4_F32.
// Grid.x = ceil(N/16) M-tiles. Block = 256 threads = 8 waves.
// Each wave computes two 16x16 N-tiles; 8 waves cover the full 256-wide strip.
// ---------------------------------------------------------------------------
__global__ __launch_bounds__(256) void gcn_gemm_support(
    const float* __restrict__ X, const float* __restrict__ W,
    float* __restrict__ S, int Nn) {
  const int m0   = blockIdx.x * 16;
  const int wave = threadIdx.x >> 5;
  const int lane = threadIdx.x & 31;
  const int ll   = lane & 15;   // lane within half-wave
  const int lh   = lane >> 4;   // which half-wave (0/1)
  const int n0   = wave * 32;   // this wave's first N-tile column

  // Clamp tail rows so every lane issues a valid load (EXEC must stay all-1s).
  int rowi = m0 + ll;
  if (rowi >= Nn) rowi = Nn - 1;
  const float* xrow = X + (size_t)rowi * F_IN;

  v8f acc0 = {};
  v8f acc1 = {};

  for (int k0 = 0; k0 < F_IN; k0 += 4) {
    // A frag: lanes 0-15 -> (M=ll, K=k0,k0+1); lanes 16-31 -> (M=ll, K=k0+2,k0+3)
    v2f a = *(const v2f*)(xrow + k0 + 2 * lh);

    // B frags: VGPR j, half h -> W[k0 + 2h + j][n0 + ll]
    const float* wp = W + (size_t)(k0 + 2 * lh) * F_OUT + ll;
    v2f b0, b1;
    b0.x = wp[n0];
    b0.y = wp[F_OUT + n0];
    b1.x = wp[n0 + 16];
    b1.y = wp[F_OUT + n0 + 16];

    acc0 = __builtin_amdgcn_wmma_f32_16x16x4_f32(
        false, a, false, b0, (short)0, acc0, false, false);
    acc1 = __builtin_amdgcn_wmma_f32_16x16x4_f32(
        false, a, false, b1, (short)0, acc1, false, false);
  }

  // D layout: VGPR r -> row m0 + r + 8*lh, col n0 + ll
#pragma unroll
  for (int r = 0; r < 8; ++r) {
    int m = m0 + r + 8 * lh;
    if (m < Nn) {
      S[(size_t)m * F_OUT + n0 + ll]      = acc0[r];
      S[(size_t)m * F_OUT + n0 + 16 + ll] = acc1[r];
    }
  }
}

// ---------------------------------------------------------------------------
// Kernel 2: COO scatter: agg[row[e]] += edge_val[e] * support[col[e]]
// One wave per edge; each lane handles 8 of the 256 features.
// support & agg both fit in the 192MB L2 -> atomics/gathers are L2-resident.
// ---------------------------------------------------------------------------
__global__ __launch_bounds__(256) void gcn_spmm_scatter(
    const float* __restrict__ S, const float* __restrict__ ev,
    const int* __restrict__ row, const int* __restrict__ col,
    float* __restrict__ agg, int E_) {
  const int e = (int)((blockIdx.x * 256u + threadIdx.x) >> 5);
  if (e >= E_) return;
  const int lane = threadIdx.x & 31;

  const int   c  = col[e];
  const int   r  = row[e];
  const float wv = ev[e];

  const float4* src = (const float4*)(S + (size_t)c * F_OUT) + lane * 2;
  float4 v0 = src[0];
  float4 v1 = src[1];

  float* dst = agg + (size_t)r * F_OUT + lane * 8;
  __hip_atomic_fetch_add(dst + 0, wv * v0.x, __ATOMIC_RELAXED, __HIP_MEMORY_SCOPE_AGENT);
  __hip_atomic_fetch_add(dst + 1, wv * v0.y, __ATOMIC_RELAXED, __HIP_MEMORY_SCOPE_AGENT);
  __hip_atomic_fetch_add(dst + 2, wv * v0.z, __ATOMIC_RELAXED, __HIP_MEMORY_SCOPE_AGENT);
  __hip_atomic_fetch_add(dst + 3, wv * v0.w, __ATOMIC_RELAXED, __HIP_MEMORY_SCOPE_AGENT);
  __hip_atomic_fetch_add(dst + 4, wv * v1.x, __ATOMIC_RELAXED, __HIP_MEMORY_SCOPE_AGENT);
  __hip_atomic_fetch_add(dst + 5, wv * v1.y, __ATOMIC_RELAXED, __HIP_MEMORY_SCOPE_AGENT);
  __hip_atomic_fetch_add(dst + 6, wv * v1.z, __ATOMIC_RELAXED, __HIP_MEMORY_SCOPE_AGENT);
  __hip_atomic_fetch_add(dst + 7, wv * v1.w, __ATOMIC_RELAXED, __HIP_MEMORY_SCOPE_AGENT);
}

// ---------------------------------------------------------------------------
// Kernel 3: in-place: out = rint(out*1000)/1000 + bias   (rint == RNE, matches
// jnp.round half-to-even). float4 vectorized; 256 % 4 == 0 so bias aligns.
// ---------------------------------------------------------------------------
__global__ __launch_bounds__(256) void gcn_finalize(
    float* __restrict__ out, const float* __restrict__ bias, long total4) {
  long i = (long)blockIdx.x * 256 + threadIdx.x;
  if (i >= total4) return;
  float4 a = ((const float4*)out)[i];
  int bofs = (int)((i * 4) & (F_OUT - 1));
  float4 b = *(const float4*)(bias + bofs);
  float4 o;
  o.x = __builtin_rintf(a.x * 1000.0f) / 1000.0f + b.x;
  o.y = __builtin_rintf(a.y * 1000.0f) / 1000.0f + b.y;
  o.z = __builtin_rintf(a.z * 1000.0f) / 1000.0f + b.z;
  o.w = __builtin_rintf(a.w * 1000.0f) / 1000.0f + b.w;
  ((float4*)out)[i] = o;
}

extern "C" void kernel_launch(void* const* d_in, const int* in_sizes, int n_in,
                              void* d_out, int out_size, void* d_ws, size_t ws_size,
                              hipStream_t stream) {
  const float* x    = (const float*)d_in[0];   // [N, 512]
  const float* w    = (const float*)d_in[1];   // [512, 256]
  const float* bias = (const float*)d_in[2];   // [256]
  const float* ev   = (const float*)d_in[3];   // [E]
  const int*   row  = (const int*)d_in[4];     // [E]
  const int*   col  = (const int*)d_in[5];     // [E]
  float*       out  = (float*)d_out;           // [N, 256]

  const int N = in_sizes[0] / F_IN;
  const int E = in_sizes[3];

  float* support = (float*)d_ws;               // N*256 f32 (~102 MB) in scratch

  // Accumulate directly into d_out; zero it first (capture-legal memset node).
  hipMemsetAsync(out, 0, (size_t)N * F_OUT * sizeof(float), stream);

  const int mtiles = (N + 15) / 16;
  gcn_gemm_support<<<mtiles, 256, 0, stream>>>(x, w, support, N);

  const int waves_blocks = (E + 7) / 8;        // 8 edges (waves) per 256-thr block
  gcn_spmm_scatter<<<waves_blocks, 256, 0, stream>>>(support, ev, row, col, out, E);

  const long total4 = (long)N * F_OUT / 4;
  const int  fblocks = (int)((total4 + 255) / 256);
  gcn_finalize<<<fblocks, 256, 0, stream>>>(out, bias, total4);
}